// MEANCURV_79096117723816
// MI455X (gfx1250) — compile-verified
//
#include <hip/hip_runtime.h>
#include <hip/hip_bf16.h>
#include <math.h>

typedef __attribute__((ext_vector_type(2))) float        v2f;
typedef __attribute__((ext_vector_type(8))) float        v8f;
typedef __attribute__((ext_vector_type(4))) unsigned int v4u;
typedef __attribute__((ext_vector_type(8))) int          v8i;
typedef __attribute__((ext_vector_type(4))) int          v4i;

#define F_DIM   128
#define ATT     64
#define HEADS   4
#define WPITCH  130                    // 128 dwords + 2 pad dwords per row (TDM pad)
#define WK_OFF  (64 * WPITCH)          // float index of Wk tile in LDS
#define LDS_BYTES (2 * 64 * WPITCH * 4)  // 66560 B

// ---------- monotonic float <-> uint32 key (for atomicMax-based segment max) ----------
__device__ __forceinline__ unsigned fkey(float f) {
    unsigned b = __float_as_uint(f);
    unsigned mask = (b & 0x80000000u) ? 0xFFFFFFFFu : 0x80000000u;
    return b ^ mask;
}
__device__ __forceinline__ float funkey(unsigned k) {
    unsigned b = (k & 0x80000000u) ? (k ^ 0x80000000u) : ~k;
    return __uint_as_float(b);
}

// ---------- zero fill ----------
__global__ void MC_zero32(unsigned* __restrict__ p, size_t n) {
    size_t i = (size_t)blockIdx.x * blockDim.x + threadIdx.x;
    if (i < n) p[i] = 0u;
}

// ---------- TDM: 2D tile (rows x 128 f32) global -> LDS, padded rows (pitch 130 dwords) ----------
// D# packing per CDNA5 ISA.  6-arg builtin form
// (clang-23 / therock toolchain): (v4u g0, v8i g1, v4i g2, v4i g3, v8i g4, i32 cpol).
__device__ __forceinline__ void tdm_load_w(unsigned ldsByteOff, const float* gptr,
                                           unsigned rows, unsigned cols, unsigned strideElems) {
    unsigned long long ga = (unsigned long long)(uintptr_t)gptr;
    v4u g0;
    g0[0] = 1u;                                          // count=1 (valid), user mode
    g0[1] = ldsByteOff;                                  // lds_addr (bytes)
    g0[2] = (unsigned)(ga & 0xFFFFFFFFu);                // global_addr[31:0]
    g0[3] = (unsigned)((ga >> 32) & 0x01FFFFFFu)         // global_addr[56:32]
          | 0x80000000u;                                 // type=2 ("image") in bits 127:126
    v8i g1;
    g1[0] = (int)((2u << 16)                             // data_size = 4 bytes
          | (1u << 20)                                   // pad_enable
          | (6u << 22)                                   // pad_interval: 128 dwords
          | (1u << 25));                                 // pad_amount: 2 dwords
    g1[1] = (int)((cols & 0xFFFFu) << 16);               // tensor_dim0[15:0] @ bits 63:48
    g1[2] = (int)(((cols >> 16) & 0xFFFFu)               // tensor_dim0[31:16]
          | ((rows & 0xFFFFu) << 16));                   // tensor_dim1[15:0]
    g1[3] = (int)(((rows >> 16) & 0xFFFFu)               // tensor_dim1[31:16]
          | ((cols & 0xFFFFu) << 16));                   // tile_dim0 = cols
    g1[4] = (int)(rows & 0xFFFFu);                       // tile_dim1 = rows, tile_dim2 = 0
    g1[5] = (int)strideElems;                            // tensor_dim0_stride[31:0]
    g1[6] = 0;                                           // stride hi / dim1_stride lo
    g1[7] = 0;
    v4i z4 = {0, 0, 0, 0};
    v8i z8 = {0, 0, 0, 0, 0, 0, 0, 0};
    __builtin_amdgcn_tensor_load_to_lds(g0, g1, z4, z4, z8, 0);
}

// ---------- Kernel 1: Q/K projection via V_WMMA_F32_16X16X4_F32 ----------
// Wq/Wk staged into LDS once per block by the Tensor Data Mover; one wave per
// 16-node tile computes q = x*Wq^T + bq and k = x*Wk^T + bk.
__global__ void MC_qk_proj_wmma(const float* __restrict__ x,
                                const float* __restrict__ Wq, const float* __restrict__ bq,
                                const float* __restrict__ Wk, const float* __restrict__ bk,
                                float* __restrict__ qout, float* __restrict__ kout,
                                int N, int numTiles) {
    extern __shared__ float ldsW[];    // [0 .. 64*130) = Wq, [WK_OFF .. ) = Wk

    // --- stage weights with TDM (wave 0 issues, everyone waits at the barrier) ---
    if (threadIdx.x < 32) {
        tdm_load_w(0u,            Wq, 64u, 128u, 128u);
        tdm_load_w(WK_OFF * 4u,   Wk, 64u, 128u, 128u);
        __builtin_amdgcn_s_wait_tensorcnt(0);
    }
    __syncthreads();

    const int waveId = (int)(((size_t)blockIdx.x * blockDim.x + threadIdx.x) >> 5);
    if (waveId >= numTiles) return;       // uniform per wave (after the barrier)

    const int lane = threadIdx.x & 31;
    const int m    = lane & 15;           // A row (node in tile) / B column
    const int kh   = lane >> 4;           // which K pair {0,1} vs {2,3}

    int node = waveId * 16 + m;
    int nrow = node < N ? node : N - 1;

    v8f accq[4] = {};
    v8f acck[4] = {};

    const float* xrow = x + (size_t)nrow * F_DIM + 2 * kh;

    for (int kb = 0; kb < F_DIM; kb += 4) {
        // load A fragment + all 8 B fragments first, then 8 back-to-back WMMAs
        v2f a;
        {
            float2 t = *(const float2*)(xrow + kb);
            a.x = t.x; a.y = t.y;
        }
        v2f bqf[4], bkf[4];
#pragma unroll
        for (int nt = 0; nt < 4; ++nt) {
            const int n = nt * 16 + m;
            float2 tq = *(const float2*)(ldsW + (size_t)n * WPITCH + kb + 2 * kh);
            float2 tk = *(const float2*)(ldsW + WK_OFF + (size_t)n * WPITCH + kb + 2 * kh);
            bqf[nt].x = tq.x; bqf[nt].y = tq.y;
            bkf[nt].x = tk.x; bkf[nt].y = tk.y;
        }
#pragma unroll
        for (int nt = 0; nt < 4; ++nt) {
            accq[nt] = __builtin_amdgcn_wmma_f32_16x16x4_f32(
                false, a, false, bqf[nt], (short)0, accq[nt], false, false);
            acck[nt] = __builtin_amdgcn_wmma_f32_16x16x4_f32(
                false, a, false, bkf[nt], (short)0, acck[nt], false, false);
        }
    }
    // C/D layout: lane, vgpr r -> element (M = r + 8*(lane>>4), Ncol = lane&15)
#pragma unroll
    for (int nt = 0; nt < 4; ++nt) {
        const int n = nt * 16 + m;
        const float biasq = bq[n];
        const float biask = bk[n];
#pragma unroll
        for (int r = 0; r < 8; ++r) {
            int outNode = waveId * 16 + r + 8 * kh;
            if (outNode < N) {
                qout[(size_t)outNode * ATT + n] = accq[nt][r] + biasq;
                kout[(size_t)outNode * ATT + n] = acck[nt][r] + biask;
            }
        }
    }
}

// ---------- Kernel 2a: per-edge per-head scores + segment max (atomicMax on keys) ----------
__global__ void MC_edge_scores(const long long* __restrict__ ei, long long E,
                               const float* __restrict__ q, const float* __restrict__ k,
                               float* __restrict__ prods, unsigned* __restrict__ segkey) {
    long long e = (long long)blockIdx.x * blockDim.x + threadIdx.x;
    if (e >= E) return;
    int row = (int)ei[e];
    int col = (int)ei[E + e];
    const float4* q4 = (const float4*)(q + (size_t)row * ATT);
    const float4* k4 = (const float4*)(k + (size_t)col * ATT);
    float s[HEADS] = {0.f, 0.f, 0.f, 0.f};
#pragma unroll
    for (int j = 0; j < 16; ++j) {        // 64 dims, head = (4*j)/16
        float4 qa = q4[j];
        float4 ka = k4[j];
        s[j >> 2] += qa.x * ka.x + qa.y * ka.y + qa.z * ka.z + qa.w * ka.w;
    }
#pragma unroll
    for (int h = 0; h < HEADS; ++h) {
        float v = s[h] * 0.25f;           // / sqrt(D_K=16)
        prods[e * HEADS + h] = v;
        atomicMax(&segkey[(size_t)row * HEADS + h], fkey(v));
    }
}

// ---------- Kernel 2b: ex = exp(score - segmax); segment sum of ex ----------
__global__ void MC_edge_exp(const long long* __restrict__ ei, long long E,
                            float* __restrict__ prods, const unsigned* __restrict__ segkey,
                            float* __restrict__ denom) {
    long long e = (long long)blockIdx.x * blockDim.x + threadIdx.x;
    if (e >= E) return;
    int row = (int)ei[e];
#pragma unroll
    for (int h = 0; h < HEADS; ++h) {
        float mx = funkey(segkey[(size_t)row * HEADS + h]);
        float ex = expf(prods[e * HEADS + h] - mx);
        prods[e * HEADS + h] = ex;
        atomicAdd(&denom[(size_t)row * HEADS + h], ex);
    }
}

// ---------- Kernel 2c: head-mean attention per edge ----------
__global__ void MC_edge_mean(const long long* __restrict__ ei, long long E,
                             const float* __restrict__ ex, const float* __restrict__ denom,
                             float* __restrict__ mean_attn) {
    long long e = (long long)blockIdx.x * blockDim.x + threadIdx.x;
    if (e >= E) return;
    int row = (int)ei[e];
    float s = 0.f;
#pragma unroll
    for (int h = 0; h < HEADS; ++h)
        s += ex[e * HEADS + h] / (denom[(size_t)row * HEADS + h] + 1e-16f);
    mean_attn[e] = 0.25f * s;
}

// ---------- Kernel 2d: agg[row] += mean_attn * x[col]   (one wave per edge) ----------
__global__ void MC_edge_agg(const long long* __restrict__ ei, long long E,
                            const float* __restrict__ x, const float* __restrict__ mean_attn,
                            float* __restrict__ agg) {
    long long t = (long long)blockIdx.x * blockDim.x + threadIdx.x;
    long long e = t >> 5;
    if (e >= E) return;
    int lane = threadIdx.x & 31;
    int row = (int)ei[e];
    int col = (int)ei[E + e];
    float a = mean_attn[e];
    float4 xv = ((const float4*)(x + (size_t)col * F_DIM))[lane];
    float* dst = agg + (size_t)row * F_DIM + lane * 4;
    atomicAdd(dst + 0, a * xv.x);
    atomicAdd(dst + 1, a * xv.y);
    atomicAdd(dst + 2, a * xv.z);
    atomicAdd(dst + 3, a * xv.w);
}

// ---------- Kernel 3: inv-norm of grad_x = agg - x   (one wave per node) ----------
__global__ void MC_node_inv(const float* __restrict__ x, const float* __restrict__ agg,
                            float* __restrict__ inv, int N) {
    long long t = (long long)blockIdx.x * blockDim.x + threadIdx.x;
    int n = (int)(t >> 5);
    if (n >= N) return;
    int lane = threadIdx.x & 31;
    float4 av = ((const float4*)(agg + (size_t)n * F_DIM))[lane];
    float4 xv = ((const float4*)(x + (size_t)n * F_DIM))[lane];
    float gx = av.x - xv.x, gy = av.y - xv.y, gz = av.z - xv.z, gw = av.w - xv.w;
    float s = fmaxf(gx * gx, 0.1f) + fmaxf(gy * gy, 0.1f) +
              fmaxf(gz * gz, 0.1f) + fmaxf(gw * gw, 0.1f);
#pragma unroll
    for (int off = 16; off > 0; off >>= 1) s += __shfl_xor(s, off, 32);
    if (lane == 0) inv[n] = 1.0f / sqrtf(s);
}

// ---------- Kernel 4a: score2 = inv[row]+inv[col]; segment max ----------
__global__ void MC_edge_score2(const long long* __restrict__ ei, long long E,
                               const float* __restrict__ inv, unsigned* __restrict__ segkey2) {
    long long e = (long long)blockIdx.x * blockDim.x + threadIdx.x;
    if (e >= E) return;
    int row = (int)ei[e];
    int col = (int)ei[E + e];
    atomicMax(&segkey2[row], fkey(inv[row] + inv[col]));
}

// ---------- Kernel 4b: ex2 = exp(score2 - max); segment sum ----------
__global__ void MC_edge_exp2(const long long* __restrict__ ei, long long E,
                             const float* __restrict__ inv, const unsigned* __restrict__ segkey2,
                             float* __restrict__ ex2, float* __restrict__ denom2) {
    long long e = (long long)blockIdx.x * blockDim.x + threadIdx.x;
    if (e >= E) return;
    int row = (int)ei[e];
    int col = (int)ei[E + e];
    float v = expf(inv[row] + inv[col] - funkey(segkey2[row]));
    ex2[e] = v;
    atomicAdd(&denom2[row], v);
}

// ---------- Kernel 5: new_attn; rowsum and ax accumulation (one wave per edge) ----------
__global__ void MC_edge_accum2(const long long* __restrict__ ei, long long E,
                               const float* __restrict__ x, const float* __restrict__ mean_attn,
                               const float* __restrict__ ex2, const float* __restrict__ denom2,
                               float* __restrict__ rowsum, float* __restrict__ ax) {
    long long t = (long long)blockIdx.x * blockDim.x + threadIdx.x;
    long long e = t >> 5;
    if (e >= E) return;
    int lane = threadIdx.x & 31;
    int row = (int)ei[e];
    int col = (int)ei[E + e];
    float na = mean_attn[e] * ex2[e] / (denom2[row] + 1e-16f);
    if (lane == 0) atomicAdd(&rowsum[row], na);
    float4 xv = ((const float4*)(x + (size_t)col * F_DIM))[lane];
    float* dst = ax + (size_t)row * F_DIM + lane * 4;
    atomicAdd(dst + 0, na * xv.x);
    atomicAdd(dst + 1, na * xv.y);
    atomicAdd(dst + 2, na * xv.z);
    atomicAdd(dst + 3, na * xv.w);
}

// ---------- Kernel 6: out = ax - rowsum * x ----------
__global__ void MC_finalize(const float* __restrict__ x, const float* __restrict__ ax,
                            const float* __restrict__ rowsum, float* __restrict__ out, int N) {
    long long i = (long long)blockIdx.x * blockDim.x + threadIdx.x;   // one float4 per thread
    if (i >= (long long)N * (F_DIM / 4)) return;
    int n = (int)(i >> 5);
    float rs = rowsum[n];
    float4 av = ((const float4*)ax)[i];
    float4 xv = ((const float4*)x)[i];
    float4 o;
    o.x = av.x - rs * xv.x;
    o.y = av.y - rs * xv.y;
    o.z = av.z - rs * xv.z;
    o.w = av.w - rs * xv.w;
    ((float4*)out)[i] = o;
}

extern "C" void kernel_launch(void* const* d_in, const int* in_sizes, int n_in,
                              void* d_out, int out_size, void* d_ws, size_t ws_size,
                              hipStream_t stream) {
    const float* x  = (const float*)d_in[0];
    const float* Wq = (const float*)d_in[1];
    const float* bq = (const float*)d_in[2];
    const float* Wk = (const float*)d_in[3];
    const float* bk = (const float*)d_in[4];
    // d_in[5], d_in[6] = Wv, bv : unused by the reference output (mix_features=0)
    const long long* ei = (const long long*)d_in[7];

    const int  N = in_sizes[0] / F_DIM;
    const long long E = (long long)in_sizes[7] / 2;

    // ---- workspace layout (bytes) ----
    char* ws = (char*)d_ws;
    size_t off = 0;
    float*    q        = (float*)(ws + off); off += (size_t)N * ATT * 4;        // 12.8 MB
    float*    k        = (float*)(ws + off); off += (size_t)N * ATT * 4;        // 12.8 MB
    float*    prods    = (float*)(ws + off); off += (size_t)E * HEADS * 4;      // 12.8 MB (scores -> ex; ex2 reuses)
    unsigned* segkey   = (unsigned*)(ws + off); off += (size_t)N * HEADS * 4;   // 0.8 MB (reused as segkey2)
    float*    denom    = (float*)(ws + off); off += (size_t)N * HEADS * 4;      // 0.8 MB (reused as denom2)
    float*    mean_at  = (float*)(ws + off); off += (size_t)E * 4;              // 3.2 MB
    float*    agg      = (float*)(ws + off); off += (size_t)N * F_DIM * 4;      // 25.6 MB (reused as ax)
    float*    inv      = (float*)(ws + off); off += (size_t)N * 4;              // 0.2 MB
    float*    rowsum   = (float*)(ws + off); off += (size_t)N * 4;              // 0.2 MB
    (void)ws_size; (void)n_in; (void)out_size;

    const int B = 256;
    auto blocks = [](long long work, int b) { return (unsigned)((work + b - 1) / b); };

    // ---- phase 0: zero segmax keys, denoms, agg ----
    MC_zero32<<<blocks((long long)N * HEADS, B), B, 0, stream>>>(segkey, (size_t)N * HEADS);
    MC_zero32<<<blocks((long long)N * HEADS, B), B, 0, stream>>>((unsigned*)denom, (size_t)N * HEADS);
    MC_zero32<<<blocks((long long)N * F_DIM, B), B, 0, stream>>>((unsigned*)agg, (size_t)N * F_DIM);

    // ---- phase 1: Q/K projection (TDM-staged weights + WMMA f32 16x16x4) ----
    const int numTiles = (N + 15) / 16;
    MC_qk_proj_wmma<<<blocks((long long)numTiles * 32, B), B, LDS_BYTES, stream>>>(
        x, Wq, bq, Wk, bk, q, k, N, numTiles);

    // ---- phase 2: per-edge scores, grouped softmax, head-mean, first diffusion ----
    MC_edge_scores<<<blocks(E, B), B, 0, stream>>>(ei, E, q, k, prods, segkey);
    MC_edge_exp   <<<blocks(E, B), B, 0, stream>>>(ei, E, prods, segkey, denom);
    MC_edge_mean  <<<blocks(E, B), B, 0, stream>>>(ei, E, prods, denom, mean_at);
    MC_edge_agg   <<<blocks(E * 32, B), B, 0, stream>>>(ei, E, x, mean_at, agg);

    // ---- phase 3: grad-norm reweighting ----
    MC_node_inv<<<blocks((long long)N * 32, B), B, 0, stream>>>(x, agg, inv, N);

    // ---- phase 4: second softmax (reuse segkey/denom[0:N], prods[0:E] as ex2, agg as ax) ----
    MC_zero32<<<blocks(N, B), B, 0, stream>>>(segkey, (size_t)N);
    MC_zero32<<<blocks(N, B), B, 0, stream>>>((unsigned*)denom, (size_t)N);
    MC_zero32<<<blocks(N, B), B, 0, stream>>>((unsigned*)rowsum, (size_t)N);
    MC_zero32<<<blocks((long long)N * F_DIM, B), B, 0, stream>>>((unsigned*)agg, (size_t)N * F_DIM);

    float* ex2 = prods;   // reuse
    MC_edge_score2<<<blocks(E, B), B, 0, stream>>>(ei, E, inv, segkey);
    MC_edge_exp2  <<<blocks(E, B), B, 0, stream>>>(ei, E, inv, segkey, ex2, denom);
    MC_edge_accum2<<<blocks(E * 32, B), B, 0, stream>>>(ei, E, x, mean_at, ex2, denom, rowsum, agg);

    // ---- phase 5: out = ax - rowsum*x ----
    MC_finalize<<<blocks((long long)N * (F_DIM / 4), B), B, 0, stream>>>(
        x, agg, rowsum, (float*)d_out, N);
}